// LinkPredictor_86191403696160
// MI455X (gfx1250) — compile-verified
//
#include <hip/hip_runtime.h>

#define NN 1024
#define DD 128

typedef __attribute__((ext_vector_type(2))) float v2f;
typedef __attribute__((ext_vector_type(8))) float v8f;

// ---------------------------------------------------------------------------
// Kernel 1: P = z @ [W1a | W1b]  via V_WMMA_F32_16X16X4_F32
//   P layout in workspace: pi = P[0 .. N*D), pjb = P[N*D .. 2*N*D) (pjb has b1 folded in)
//   grid = (64 m-tiles, 16 n-tiles), block = 32 (one wave per 16x16 output tile)
// A layout (16x4 f32): lane l holds A[l%16][2*(l/16) + p] in vgpr p (p=0,1)
// B layout (4x16 f32): lane l holds B[2*(l/16) + p][l%16]  in vgpr p
// C layout (16x16 f32): lane l, vgpr r holds C[r + 8*(l/16)][l%16]
// ---------------------------------------------------------------------------
__global__ __launch_bounds__(32) void proj_wmma_f32(
    const float* __restrict__ z,   // [1024, 128]
    const float* __restrict__ W1,  // [256, 128]
    const float* __restrict__ b1,  // [128]
    float* __restrict__ P)         // ws: [2, 1024, 128]
{
    const int mt   = blockIdx.x;        // 0..63
    const int nt   = blockIdx.y;        // 0..15  (16 cols of the 256-wide output)
    const int lane = threadIdx.x;       // 0..31
    const int half = lane >> 4;         // 0/1
    const int l16  = lane & 15;
    const int m0   = mt * 16;
    const int c    = nt * 16 + l16;     // output col in [0,256); nt*16 never straddles 128
    // column c of [W1a | W1b]: c<128 -> W1[k][c], c>=128 -> W1[k+128][c-128]
    const int wcol  = (c < 128) ? c : (c - 128);
    const int wrow0 = (c < 128) ? 0 : 128;

    v8f acc = {};
#pragma unroll
    for (int k = 0; k < DD; k += 4) {
        const int kk = k + 2 * half;
        const float* za = z + (m0 + l16) * DD + kk;
        v2f a; a.x = za[0]; a.y = za[1];
        v2f b;
        b.x = W1[(wrow0 + kk + 0) * DD + wcol];
        b.y = W1[(wrow0 + kk + 1) * DD + wcol];
        acc = __builtin_amdgcn_wmma_f32_16x16x4_f32(
            /*neg_a=*/false, a, /*neg_b=*/false, b,
            /*c_mod=*/(short)0, acc, /*reuse_a=*/false, /*reuse_b=*/false);
    }

    // store; fold b1 into the pj half (pjb = pj + b1)
    const float bias    = (c >= 128) ? b1[wcol] : 0.0f;
    float* dst          = P + ((c < 128) ? 0 : (NN * DD));
#pragma unroll
    for (int r = 0; r < 8; ++r) {
        const int m = m0 + r + 8 * half;
        dst[m * DD + wcol] = acc[r] + bias;
    }
}

// ---------------------------------------------------------------------------
// Kernel 2: q[i,j] = sum_d relu(pi[i,d] + pjb[j,d]) * W2[d] + b2
//   block = 256 threads = 8 waves; block covers 8 i's (pi tile in LDS) and
//   8*32 = 256 j's (one j per lane per wave).
//   grid = (4 j-groups, 128 i-tiles) -> 1024 x 1024 pairs.
// ---------------------------------------------------------------------------
__global__ __launch_bounds__(256) void pairwise_score(
    const float* __restrict__ P,   // ws: pi then pjb
    const float* __restrict__ W2,  // [128]
    const float* __restrict__ b2,  // [1]
    float* __restrict__ out)       // [1024*1024], out[i*N + j]
{
    const float* pi  = P;
    const float* pjb = P + (size_t)NN * DD;

    __shared__ __align__(16) float s_pi[8][DD];  // 4 KB
    __shared__ __align__(16) float s_w[DD];      // 512 B

    const int tid = threadIdx.x;
    const int i0  = blockIdx.y * 8;

    // cooperative LDS fill: 8*128 = 1024 floats = 256 float4 (one per thread)
    {
        const float4* src = (const float4*)(pi + (size_t)i0 * DD);
        ((float4*)&s_pi[0][0])[tid] = src[tid];
        if (tid < DD / 4) ((float4*)s_w)[tid] = ((const float4*)W2)[tid];
    }
    __syncthreads();

    const int wave = tid >> 5;
    const int lane = tid & 31;
    const int j    = (blockIdx.x * 8 + wave) * 32 + lane;

    const float4* pjrow = (const float4*)(pjb + (size_t)j * DD);

    float acc[8];
#pragma unroll
    for (int r = 0; r < 8; ++r) acc[r] = 0.0f;

    for (int cidx = 0; cidx < DD / 4; ++cidx) {   // 32 chunks of 4 d's
        const float4 pj4 = pjrow[cidx];                    // per-lane, streamed once (L2-hot)
        const float4 w4  = ((const float4*)s_w)[cidx];     // LDS broadcast
#pragma unroll
        for (int r = 0; r < 8; ++r) {
            const float4 p4 = ((const float4*)&s_pi[r][0])[cidx];  // LDS broadcast
            float h0 = fmaxf(p4.x + pj4.x, 0.0f);
            float h1 = fmaxf(p4.y + pj4.y, 0.0f);
            float h2 = fmaxf(p4.z + pj4.z, 0.0f);
            float h3 = fmaxf(p4.w + pj4.w, 0.0f);
            acc[r] = fmaf(h0, w4.x, acc[r]);
            acc[r] = fmaf(h1, w4.y, acc[r]);
            acc[r] = fmaf(h2, w4.z, acc[r]);
            acc[r] = fmaf(h3, w4.w, acc[r]);
        }
    }

    const float bb = b2[0];
#pragma unroll
    for (int r = 0; r < 8; ++r)
        out[(size_t)(i0 + r) * NN + j] = acc[r] + bb;
}

// ---------------------------------------------------------------------------
extern "C" void kernel_launch(void* const* d_in, const int* in_sizes, int n_in,
                              void* d_out, int out_size, void* d_ws, size_t ws_size,
                              hipStream_t stream) {
    const float* z  = (const float*)d_in[0];   // [1024,128]
    const float* W1 = (const float*)d_in[1];   // [256,128]
    const float* b1 = (const float*)d_in[2];   // [128]
    const float* W2 = (const float*)d_in[3];   // [128,1]
    const float* b2 = (const float*)d_in[4];   // [1]
    float* out = (float*)d_out;                // [1024*1024]
    float* P   = (float*)d_ws;                 // 2*1024*128 floats = 1 MB

    (void)in_sizes; (void)n_in; (void)out_size; (void)ws_size;

    dim3 g1(NN / 16, 256 / 16);   // 64 x 16
    proj_wmma_f32<<<g1, 32, 0, stream>>>(z, W1, b1, P);

    dim3 g2(NN / (8 * 32), NN / 8);  // 4 x 128
    pairwise_score<<<g2, 256, 0, stream>>>(P, W2, b2, out);
}